// HOPInteractionLayer_36558761624331
// MI455X (gfx1250) — compile-verified
//
#include <hip/hip_runtime.h>
#include <math.h>

// ---------------------------------------------------------------------------
// HOP interaction layer, fused, fp32 WMMA (V_WMMA_F32_16X16X4_F32) on gfx1250.
//   phase 1: per-atom envsum as chunked rank-4 outer products (WMMA K=4)
//   phase 2: block GEMM env_rs(32x512) @ w_rs(512x32) (WMMA, K-chunks of 4)
//   phase 3: invariants + GroupNorm + mixing + self-interaction (VALU/shuffle)
// ---------------------------------------------------------------------------

typedef float v2f __attribute__((ext_vector_type(2)));
typedef float v8f __attribute__((ext_vector_type(8)));

#define N_ATOMS    8000
#define N_PAIRS    80000
#define NF_IN      32
#define NF_OUT     32
#define N_DIST     16
#define N_COMP     4
#define N_INV      2
#define HARD_CUT   6.5f
#define GN_EPS     1e-5f

#define APB        8                 // atoms per block
#define THREADS    128               // 4 wave32
#define ROWS       (APB * N_COMP)    // 32 rows of env_rs per block
#define KDIM       (N_DIST * NF_IN)  // 512
#define EPITCH     516               // env row pitch (floats); 516%64==4 -> conflict-free A reads

static constexpr int ENV_F = ROWS * EPITCH;           // 16512
static constexpr int W_F   = KDIM * NF_OUT;           // 16384
static constexpr int TF_F  = ROWS * NF_OUT;           // 1024
static constexpr int MW_F  = NF_OUT * N_INV * NF_OUT; // 2048
static constexpr int SW_F  = NF_OUT * NF_IN;          // 1024
static constexpr int LDS_FLOATS = ENV_F + W_F + TF_F + MW_F + SW_F + 32 + 64 + 64;

__device__ __forceinline__ int lower_bound_pf(const int* __restrict__ pf, int val) {
    int lo = 0, hi = N_PAIRS;
    while (lo < hi) {
        int mid = (lo + hi) >> 1;
        if (pf[mid] < val) lo = mid + 1; else hi = mid;
    }
    return lo;
}

__device__ __forceinline__ float wave_sum32(float x) {
    x += __shfl_xor(x, 1,  32);
    x += __shfl_xor(x, 2,  32);
    x += __shfl_xor(x, 4,  32);
    x += __shfl_xor(x, 8,  32);
    x += __shfl_xor(x, 16, 32);
    return x;
}

__global__ __launch_bounds__(THREADS)
void hop_fused(const float* __restrict__ feat,      // (N_ATOMS, NF_IN)
               const float* __restrict__ rhats,     // (N_PAIRS, N_COMP)
               const float* __restrict__ dist,      // (N_PAIRS)
               const float* __restrict__ iw,        // (N_DIST, NF_OUT, NF_IN)
               const float* __restrict__ sw,        // (NF_OUT, NF_IN)
               const float* __restrict__ sb,        // (NF_OUT)
               const float* __restrict__ mw,        // (NF_OUT, N_INV, NF_OUT)
               const float* __restrict__ gnw,       // (64)
               const float* __restrict__ gnb,       // (64)
               const float* __restrict__ mu,        // (N_DIST)
               const float* __restrict__ sg,        // (N_DIST)
               const int*   __restrict__ pfirst,    // (N_PAIRS) sorted
               const int*   __restrict__ psecond,   // (N_PAIRS)
               float*       __restrict__ out)       // (N_ATOMS, NF_OUT)
{
    extern __shared__ float lds[];
    float* envL = lds;                  // ENV_F : env_rs rows, padded pitch
    float* wL   = envL + ENV_F;         // W_F   : w_rs[k, o] = iw[k>>5, o, k&31]
    float* tfL  = wL   + W_F;           // TF_F  : tf rows (a_local*4+c, o)
    float* mwL  = tfL  + TF_F;          // MW_F
    float* swL  = mwL  + MW_F;          // SW_F
    float* sbL  = swL  + SW_F;          // 32
    float* gnwL = sbL  + 32;            // 64
    float* gnbL = gnwL + 64;            // 64

    const int tid   = threadIdx.x;
    const int wave  = tid >> 5;
    const int lane  = tid & 31;
    const int l15   = lane & 15;
    const int koff  = (lane < 16) ? 0 : 2;   // A/B K-slot base for this half-wave
    const int mbase = (lane < 16) ? 0 : 8;   // C/D row offset for this half-wave

    // ---- stage weights into LDS ----
    for (int i = tid; i < W_F; i += THREADS) {
        const int k = i >> 5, o = i & 31;                       // k = d*32 + f
        wL[i] = iw[(k >> 5) * (NF_OUT * NF_IN) + o * NF_IN + (k & 31)];
    }
    for (int i = tid; i < MW_F; i += THREADS) mwL[i] = mw[i];
    for (int i = tid; i < SW_F; i += THREADS) swL[i] = sw[i];
    if (tid < 32) sbL[tid] = sb[tid];
    if (tid < 64) { gnwL[tid] = gnw[tid]; gnbL[tid] = gnb[tid]; }

    const float mu_l  = mu[l15];
    const float isg_l = 1.0f / sg[l15];
    const float cutk  = 0.5f * 3.14159265358979323846f / HARD_CUT;
    const int   ablk  = blockIdx.x * APB;

    // ---- phase 1: envsum.  env(64x32) per atom = SenseT(64xP) @ FeatJ(Px32) ----
    for (int al = wave * 2; al < wave * 2 + 2; ++al) {
        const int a = ablk + al;
        v8f acc[N_COMP][2];
        #pragma unroll
        for (int c = 0; c < N_COMP; ++c) {
            acc[c][0] = {0.f,0.f,0.f,0.f,0.f,0.f,0.f,0.f};
            acc[c][1] = {0.f,0.f,0.f,0.f,0.f,0.f,0.f,0.f};
        }
        const int rs = lower_bound_pf(pfirst, a);
        const int re = lower_bound_pf(pfirst, a + 1);

        for (int p0 = rs; p0 < re; p0 += 4) {         // wave-uniform: EXEC stays all-1s
            const int  pA = p0 + koff, pB = pA + 1;
            const bool vA = pA < re,   vB = pB < re;
            const int  pAc = vA ? pA : rs, pBc = vB ? pB : rs;

            const float dA = dist[pAc], dB = dist[pBc];
            const float zA = (1.0f / dA - mu_l) * isg_l;
            const float zB = (1.0f / dB - mu_l) * isg_l;
            float cA = __cosf(cutk * dA); cA *= cA;
            float cB = __cosf(cutk * dB); cB *= cB;
            float sA = __expf(-0.5f * zA * zA) * ((dA < HARD_CUT) ? cA : 0.0f);
            float sB = __expf(-0.5f * zB * zB) * ((dB < HARD_CUT) ? cB : 0.0f);
            sA = vA ? sA : 0.0f;                      // zero A => tail pairs contribute 0
            sB = vB ? sB : 0.0f;

            const float4 rA = *(const float4*)(rhats + pAc * 4);
            const float4 rB = *(const float4*)(rhats + pBc * 4);
            const float  ra[4] = {rA.x, rA.y, rA.z, rA.w};
            const float  rb[4] = {rB.x, rB.y, rB.z, rB.w};

            const float* fA = feat + psecond[pAc] * NF_IN;
            const float* fB = feat + psecond[pBc] * NF_IN;
            v2f b0, b1;                                // B: vgpr0 = K=koff, vgpr1 = K=koff+1
            b0.x = fA[l15];      b0.y = fB[l15];
            b1.x = fA[16 + l15]; b1.y = fB[16 + l15];

            #pragma unroll
            for (int c = 0; c < N_COMP; ++c) {
                v2f av; av.x = sA * ra[c]; av.y = sB * rb[c];   // A[m=l15, k]
                acc[c][0] = __builtin_amdgcn_wmma_f32_16x16x4_f32(
                    false, av, false, b0, (short)0, acc[c][0], false, false);
                acc[c][1] = __builtin_amdgcn_wmma_f32_16x16x4_f32(
                    false, av, false, b1, (short)0, acc[c][1], false, false);
            }
        }
        // spill env tiles to LDS in env_rs layout: row = al*4+c, col = d*32 + f
        #pragma unroll
        for (int c = 0; c < N_COMP; ++c) {
            #pragma unroll
            for (int ft = 0; ft < 2; ++ft) {
                const v8f t = acc[c][ft];
                #pragma unroll
                for (int v = 0; v < 8; ++v)
                    envL[(al * N_COMP + c) * EPITCH + (v + mbase) * NF_IN + ft * 16 + l15] = t[v];
            }
        }
    }
    __syncthreads();

    // ---- phase 2: tf_block(32x32) = env_rs(32x512) @ w_rs(512x32) ----
    {
        const int rt = wave >> 1, ct = wave & 1;      // 4 waves = 4 output tiles
        v8f dacc = {0.f,0.f,0.f,0.f,0.f,0.f,0.f,0.f};
        const float* aRow = envL + (rt * 16 + l15) * EPITCH;  // A rows m = l15
        const float* bCol = wL + ct * 16 + l15;               // B cols n = l15
        for (int kk = 0; kk < KDIM / 4; ++kk) {
            const int k0 = kk * 4 + koff;
            const v2f av = *(const v2f*)(aRow + k0);          // b64 DS load, conflict-free
            v2f bv; bv.x = bCol[k0 * NF_OUT]; bv.y = bCol[(k0 + 1) * NF_OUT];
            dacc = __builtin_amdgcn_wmma_f32_16x16x4_f32(
                false, av, false, bv, (short)0, dacc, false, false);
        }
        #pragma unroll
        for (int v = 0; v < 8; ++v)
            tfL[(rt * 16 + v + mbase) * NF_OUT + ct * 16 + l15] = dacc[v];
    }
    __syncthreads();

    // ---- phase 3: invariants + GroupNorm + mixing + self-interaction ----
    for (int al = wave * 2; al < wave * 2 + 2; ++al) {
        const int a = ablk + al;
        const float t0 = tfL[(al * 4 + 0) * NF_OUT + lane];
        const float t1 = tfL[(al * 4 + 1) * NF_OUT + lane];
        const float t2 = tfL[(al * 4 + 2) * NF_OUT + lane];
        const float t3 = tfL[(al * 4 + 3) * NF_OUT + lane];
        const float inv0 = t0;                              // l=0 scalar
        const float inv1 = t1 * t1 + t2 * t2 + t3 * t3;     // |l=1 vector|^2

        const float m0 = wave_sum32(inv0)        * (1.0f / 32.0f);
        const float q0 = wave_sum32(inv0 * inv0) * (1.0f / 32.0f);
        const float m1 = wave_sum32(inv1)        * (1.0f / 32.0f);
        const float q1 = wave_sum32(inv1 * inv1) * (1.0f / 32.0f);
        const float xn0 = (inv0 - m0) * rsqrtf(q0 - m0 * m0 + GN_EPS) * gnwL[lane]      + gnbL[lane];
        const float xn1 = (inv1 - m1) * rsqrtf(q1 - m1 * m1 + GN_EPS) * gnwL[32 + lane] + gnbL[32 + lane];

        float accout = sbL[lane];                           // lane = output channel
        const float fmine = feat[a * NF_IN + lane];
        #pragma unroll 8
        for (int o = 0; o < 32; ++o) {
            const float x0 = __shfl(xn0,   o, 32);
            const float x1 = __shfl(xn1,   o, 32);
            const float ff = __shfl(fmine, o, 32);
            accout += x0 * mwL[o * 64 + lane]
                    + x1 * mwL[o * 64 + 32 + lane]
                    + ff * swL[lane * NF_IN + o];
        }
        out[a * NF_OUT + lane] = accout;
    }
}

extern "C" void kernel_launch(void* const* d_in, const int* in_sizes, int n_in,
                              void* d_out, int out_size, void* d_ws, size_t ws_size,
                              hipStream_t stream) {
    (void)in_sizes; (void)n_in; (void)out_size; (void)d_ws; (void)ws_size;
    const float* feat    = (const float*)d_in[0];
    const float* rhats   = (const float*)d_in[1];
    const float* dist    = (const float*)d_in[2];
    const float* iw      = (const float*)d_in[3];
    const float* sw      = (const float*)d_in[4];
    const float* sb      = (const float*)d_in[5];
    const float* mw      = (const float*)d_in[6];
    const float* gnw     = (const float*)d_in[7];
    const float* gnb     = (const float*)d_in[8];
    const float* mu      = (const float*)d_in[9];
    const float* sg      = (const float*)d_in[10];
    const int*   pfirst  = (const int*)d_in[11];
    const int*   psecond = (const int*)d_in[12];
    float*       out     = (float*)d_out;

    const size_t ldsBytes = (size_t)LDS_FLOATS * sizeof(float);  // ~148.6 KB < 320 KB WGP
    hop_fused<<<N_ATOMS / APB, THREADS, ldsBytes, stream>>>(
        feat, rhats, dist, iw, sw, sb, mw, gnw, gnb, mu, sg, pfirst, psecond, out);
}